// AttentionSumLayer_38156489458110
// MI455X (gfx1250) — compile-verified
//
#include <hip/hip_runtime.h>
#include <math.h>

#define D 64
#define NEG_SLOPE 0.2f

typedef float v2f __attribute__((ext_vector_type(2)));
typedef float v8f __attribute__((ext_vector_type(8)));

__device__ __forceinline__ float leaky(float v) { return v > 0.f ? v : NEG_SLOPE * v; }

// Float atomic max via monotone integer bit trick (exact, works for mixed signs):
// non-negative floats: signed-int max; negative floats: unsigned-int min.
__device__ __forceinline__ void atomicMaxF32(float* addr, float val) {
  if (val >= 0.f) {
    atomicMax((int*)addr, __float_as_int(val));
  } else {
    atomicMin((unsigned int*)addr, (unsigned int)__float_as_int(val));
  }
}

// ---------------------------------------------------------------- init
__global__ void k_init(float* __restrict__ out, float* __restrict__ m,
                       float* __restrict__ s, int N, long total) {
  long i = (long)blockIdx.x * blockDim.x + threadIdx.x;
  if (i < total) out[i] = 0.f;
  if (i < N) { m[i] = -INFINITY; s[i] = 0.f; }
}

// ---------------------------------------------------------------- z = x @ W^T + b
// Wave computes a 16-row stripe with V_WMMA_F32_16X16X4_F32.
// A 16x4 layout: lanes 0-15 -> M=lane, K={0,1}; lanes 16-31 -> M=lane-16, K={2,3}
// B 4x16 layout: lanes 0-15 -> N=lane, K={0,1}; lanes 16-31 -> N=lane-16, K={2,3}
// C 16x16 layout: VGPR r -> row r (lanes 0-15) / row r+8 (lanes 16-31), col = lane%16
__launch_bounds__(256)
__global__ void k_gemm(const float* __restrict__ x, const float* __restrict__ W,
                       const float* __restrict__ bias, float* __restrict__ z, int N) {
  __shared__ float Wsh[64 * 66];   // stride-66 pad: conflict-free column reads, 8B aligned
  __shared__ float bsh[64];

  for (int idx = threadIdx.x; idx < 64 * 64; idx += 256) {
    int r = idx >> 6, c = idx & 63;
    Wsh[r * 66 + c] = W[idx];      // W row-major [out][in]
  }
  if (threadIdx.x < 64) bsh[threadIdx.x] = bias[threadIdx.x];
  __syncthreads();

  const int wave = threadIdx.x >> 5;
  const int lane = threadIdx.x & 31;
  const int half = lane >> 4;         // K pair select (+0 or +2)
  const int l16  = lane & 15;
  const int rowBase = blockIdx.x * 128 + wave * 16;

  int arow = rowBase + l16;
  if (arow >= N) arow = N - 1;        // clamp loads; stores are guarded
  const float* xr = x + (size_t)arow * D;

  v2f afrag[16];
  #pragma unroll
  for (int ks = 0; ks < 16; ++ks) {
    float2 t = *(const float2*)(xr + 4 * ks + 2 * half);
    afrag[ks].x = t.x; afrag[ks].y = t.y;
  }

  #pragma unroll
  for (int nt = 0; nt < 4; ++nt) {
    const int col = nt * 16 + l16;
    const float* wrow = &Wsh[col * 66];
    v8f acc = {};
    #pragma unroll
    for (int ks = 0; ks < 16; ++ks) {
      float2 t = *(const float2*)(wrow + 4 * ks + 2 * half);  // B[k][col] = W[col][k]
      v2f bfrag; bfrag.x = t.x; bfrag.y = t.y;
      acc = __builtin_amdgcn_wmma_f32_16x16x4_f32(
          false, afrag[ks], false, bfrag, (short)0, acc, false, false);
    }
    float bb = bsh[col];
    #pragma unroll
    for (int r = 0; r < 8; ++r) {
      int orow = rowBase + r + 8 * half;
      if (orow < N) z[(size_t)orow * D + col] = acc[r] + bb;
    }
  }
}

// ---------------------------------------------------------------- e + segment max
__launch_bounds__(256)
__global__ void k_edge_score(const int* __restrict__ src, const int* __restrict__ dst,
                             const float* __restrict__ z, const float* __restrict__ att,
                             float* __restrict__ e, float* __restrict__ m, int E) {
  int edge = (blockIdx.x * 256 + threadIdx.x) >> 5;
  if (edge >= E) return;
  int lane = threadIdx.x & 31;
  int sj = src[edge], di = dst[edge];
  float2 zs = *(const float2*)(z + (size_t)sj * D + lane * 2);
  float2 zd = *(const float2*)(z + (size_t)di * D + lane * 2);
  float2 av = *(const float2*)(att + lane * 2);
  float v = leaky(zs.x + zd.x) * av.x + leaky(zs.y + zd.y) * av.y;
  #pragma unroll
  for (int off = 16; off > 0; off >>= 1) v += __shfl_xor(v, off, 32);
  if (lane == 0) {
    e[edge] = v;
    atomicMaxF32(&m[di], v);
  }
}

// ---------------------------------------------------------------- segment sum of exp
__launch_bounds__(256)
__global__ void k_expsum(const int* __restrict__ dst, const float* __restrict__ e,
                         const float* __restrict__ m, float* __restrict__ s, int E) {
  int i = blockIdx.x * 256 + threadIdx.x;
  if (i >= E) return;
  int di = dst[i];
  atomicAdd(&s[di], expf(e[i] - m[di]));
}

// ---------------------------------------------------------------- weighted scatter-add
__launch_bounds__(256)
__global__ void k_aggregate(const int* __restrict__ src, const int* __restrict__ dst,
                            const float* __restrict__ x, const float* __restrict__ e,
                            const float* __restrict__ m, const float* __restrict__ s,
                            float* __restrict__ out, int E) {
  int edge = (blockIdx.x * 256 + threadIdx.x) >> 5;
  if (edge >= E) return;
  int lane = threadIdx.x & 31;
  int sj = src[edge], di = dst[edge];
  float alpha = expf(e[edge] - m[di]) / s[di];
  float2 xv = *(const float2*)(x + (size_t)sj * D + lane * 2);
  float* o = out + (size_t)di * D + lane * 2;
  atomicAdd(o, alpha * xv.x);
  atomicAdd(o + 1, alpha * xv.y);
}

// ---------------------------------------------------------------- launcher
extern "C" void kernel_launch(void* const* d_in, const int* in_sizes, int n_in,
                              void* d_out, int out_size, void* d_ws, size_t ws_size,
                              hipStream_t stream) {
  const float* x   = (const float*)d_in[0];
  const int*   ei  = (const int*)  d_in[1];
  const float* W   = (const float*)d_in[2];
  const float* b   = (const float*)d_in[3];
  const float* att = (const float*)d_in[4];

  const int N = in_sizes[0] / D;
  const int E = in_sizes[1] / 2;
  const int* src = ei;        // edge_index[0]
  const int* dst = ei + E;    // edge_index[1]
  float* out = (float*)d_out;

  char* ws = (char*)d_ws;
  float* z = (float*)ws; ws += (size_t)N * D * sizeof(float);
  float* e = (float*)ws; ws += (size_t)E * sizeof(float);
  float* m = (float*)ws; ws += (size_t)N * sizeof(float);
  float* s = (float*)ws;

  long total = (long)N * D;
  k_init<<<(int)((total + 255) / 256), 256, 0, stream>>>(out, m, s, N, total);

  k_gemm<<<(N + 127) / 128, 256, 0, stream>>>(x, W, b, z, N);

  long et = (long)E * 32;
  k_edge_score<<<(int)((et + 255) / 256), 256, 0, stream>>>(src, dst, z, att, e, m, E);

  k_expsum<<<(E + 255) / 256, 256, 0, stream>>>(dst, e, m, s, E);

  k_aggregate<<<(int)((et + 255) / 256), 256, 0, stream>>>(src, dst, x, e, m, s, out, E);
}